// LModel14_71837622993175
// MI455X (gfx1250) — compile-verified
//
#include <hip/hip_runtime.h>
#include <math.h>

#define EPSF 1e-6f
#define IMG_H 1500
#define IMG_W 2000
#define SAMPLE_BLOCKS 1024
#define SAMPLE_THREADS 256
#define EDGE_STRIDE 20   // floats per edge record

typedef float v2f __attribute__((ext_vector_type(2)));
typedef float v8f __attribute__((ext_vector_type(8)));

// ---------- small vector helpers ----------
__device__ __forceinline__ void cross3(const float* a, const float* b, float* o) {
    o[0] = a[1]*b[2] - a[2]*b[1];
    o[1] = a[2]*b[0] - a[0]*b[2];
    o[2] = a[0]*b[1] - a[1]*b[0];
}
__device__ __forceinline__ float dot3(const float* a, const float* b) {
    return a[0]*b[0] + a[1]*b[1] + a[2]*b[2];
}
// reference _normalize: x / (||x|| + EPS)
__device__ __forceinline__ void normalize3(float* v) {
    float n = sqrtf(dot3(v, v)) + EPSF;
    v[0] /= n; v[1] /= n; v[2] /= n;
}

// ---------- kernel 1: T = K2h @ E2 @ inv(E1)  (single thread) ----------
__global__ void precompute_T(const float* __restrict__ K2,
                             const float* __restrict__ E1,
                             const float* __restrict__ E2,
                             float* __restrict__ T) {
    if (threadIdx.x != 0 || blockIdx.x != 0) return;
    double a[4][8];
    for (int r = 0; r < 4; ++r)
        for (int c = 0; c < 4; ++c) {
            a[r][c]     = (double)E1[r*4 + c];
            a[r][c + 4] = (r == c) ? 1.0 : 0.0;
        }
    // Gauss-Jordan with partial pivoting
    for (int col = 0; col < 4; ++col) {
        int piv = col;
        for (int r = col + 1; r < 4; ++r)
            if (fabs(a[r][col]) > fabs(a[piv][col])) piv = r;
        if (piv != col)
            for (int c = 0; c < 8; ++c) { double t = a[col][c]; a[col][c] = a[piv][c]; a[piv][c] = t; }
        double d = a[col][col];
        for (int c = 0; c < 8; ++c) a[col][c] /= d;
        for (int r = 0; r < 4; ++r) if (r != col) {
            double f = a[r][col];
            for (int c = 0; c < 8; ++c) a[r][c] -= f * a[col][c];
        }
    }
    double M[4][4];
    for (int r = 0; r < 4; ++r)
        for (int c = 0; c < 4; ++c) {
            double s = 0.0;
            for (int k = 0; k < 4; ++k) s += (double)E2[r*4 + k] * a[k][c + 4];
            M[r][c] = s;
        }
    // rows 0..2: K2 @ M ; row 3: M row 3
    for (int r = 0; r < 3; ++r)
        for (int c = 0; c < 4; ++c) {
            double s = 0.0;
            for (int k = 0; k < 3; ++k) s += (double)K2[r*3 + k] * M[k][c];
            T[r*4 + c] = (float)s;
        }
    for (int c = 0; c < 4; ++c) T[12 + c] = (float)M[3][c];
}

// ---------- kernel 2: per-edge geometry, folded projection matrices, loss terms ----------
__global__ void edge_kernel(const float* __restrict__ ep,   // (N,4,3)
                            const float* __restrict__ K1,   // 3x3
                            const float* __restrict__ T,    // 4x4 (rows 0..2 used)
                            float* __restrict__ edgeF,      // N * EDGE_STRIDE
                            int*   __restrict__ numh,
                            float* __restrict__ nl,
                            float* __restrict__ sn,
                            int N) {
    int e = blockIdx.x * blockDim.x + threadIdx.x;
    if (e >= N) return;
    const float* P = ep + (size_t)e * 12;
    float p0[3] = {P[0], P[1], P[2]};
    float p1[3] = {P[3], P[4], P[5]};
    float p2[3] = {P[6], P[7], P[8]};
    float p3[3] = {P[9], P[10], P[11]};

    float cd[3] = {p1[0]-p0[0], p1[1]-p0[1], p1[2]-p0[2]};
    // cl = ||cd + EPS||  (EPS added per-component, per reference)
    float e0 = cd[0] + EPSF, e1 = cd[1] + EPSF, e2 = cd[2] + EPSF;
    float cl = sqrtf(e0*e0 + e1*e1 + e2*e2);

    int nh = (int)floorf(cl / 0.05f);
    nh = nh < 2 ? 2 : (nh > 1000 ? 1000 : nh);
    numh[e] = nh;

    float dir[3] = {cd[0]/cl, cd[1]/cl, cd[2]/cl};
    float nxt[3] = {p3[0]-p1[0], p3[1]-p1[1], p3[2]-p1[2]};
    float prv[3] = {p0[0]-p2[0], p0[1]-p2[1], p0[2]-p2[2]};

    float cn[3]; cross3(dir, nxt, cn); normalize3(cn);
    if (cn[2] > 0.0f) { cn[0] = -cn[0]; cn[1] = -cn[1]; cn[2] = -cn[2]; }
    float up[3]; cross3(cn, dir, up); normalize3(up);
    float pn[3]; cross3(prv, dir, pn); normalize3(pn);

    nl[e] = 1.0f - dot3(cn, pn);

    float obs[3]; cross3(p0, p1, obs); normalize3(obs);
    float snp = fabsf(dot3(up, obs));
    snp = fminf(snp, 0.5f);
    sn[e] = 1.0f - snp * 2.0f;

    float* E = edgeF + (size_t)e * EDGE_STRIDE;
    // A1 = K1 @ dir, B1 = K1 @ up, C1 = K1 @ p0
    for (int r = 0; r < 3; ++r) {
        const float* k = K1 + r*3;
        E[r]     = k[0]*dir[0] + k[1]*dir[1] + k[2]*dir[2];
        E[3 + r] = k[0]*up[0]  + k[1]*up[1]  + k[2]*up[2];
        E[6 + r] = k[0]*p0[0]  + k[1]*p0[1]  + k[2]*p0[2];
    }
    // A2/B2/C2 from T rows 0..2 (row 3 of uv2h never used by reference)
    for (int r = 0; r < 3; ++r) {
        const float* t = T + r*4;
        E[9  + r] = t[0]*dir[0] + t[1]*dir[1] + t[2]*dir[2];
        E[12 + r] = t[0]*up[0]  + t[1]*up[1]  + t[2]*up[2];
        E[15 + r] = t[0]*p0[0]  + t[1]*p0[1]  + t[2]*p0[2] + t[3];
    }
    E[18] = cl / (float)(nh - 1);   // coords_x scale
    E[19] = 0.0f;
}

// ---------- kernel 3: single-block exclusive scan of num_h ----------
__global__ void scan_kernel(const int* __restrict__ numh,
                            int* __restrict__ begin,
                            int* __restrict__ total,
                            int N) {
    __shared__ int sdata[1024];
    __shared__ int s_off;
    int t = threadIdx.x;
    if (t == 0) s_off = 0;
    __syncthreads();
    for (int base = 0; base < N; base += 1024) {
        int i = base + t;
        int v = (i < N) ? numh[i] : 0;
        sdata[t] = v;
        __syncthreads();
        for (int d = 1; d < 1024; d <<= 1) {
            int x = (t >= d) ? sdata[t - d] : 0;
            __syncthreads();
            sdata[t] += x;
            __syncthreads();
        }
        int incl = sdata[t];
        int off = s_off;
        if (i < N) begin[i] = off + incl - v;
        __syncthreads();
        if (t == 1023) s_off = off + sdata[1023];
        __syncthreads();
    }
    if (t == 0) *total = s_off;
}

// ---------- bilinear sampler (channel-planar image, 3 x H x W) ----------
// u,v in [0, 0.999999] => x < W-1, y < H-1, so x0 <= W-2 and x1 = x0+1 is
// always in range (no clamps needed; matches reference's clip semantics).
__device__ __forceinline__ void sample3(const float* __restrict__ img,
                                        float u, float v, float out[3]) {
    float x = u * (float)(IMG_W - 1);
    float y = v * (float)(IMG_H - 1);
    float xf = floorf(x), yf = floorf(y);
    int x0 = (int)xf, y0 = (int)yf;
    float wx = x - xf, wy = y - yf;
    float w00 = (1.0f - wx) * (1.0f - wy);
    float w10 = wx * (1.0f - wy);
    float w01 = (1.0f - wx) * wy;
    float w11 = wx * wy;
    int b00 = y0 * IMG_W + x0;       // row y0, cols x0 / x0+1
    int b01 = b00 + IMG_W;           // row y0+1
    const int plane = IMG_H * IMG_W;
#pragma unroll
    for (int c = 0; c < 3; ++c) {
        const float* p = img + c * plane;
        out[c] = p[b00]*w00 + p[b00 + 1]*w10 + p[b01]*w01 + p[b01 + 1]*w11;
    }
}

__device__ __forceinline__ float clipdiv(float num, float den) {
    float r = num / den;
    return fminf(fmaxf(r, 0.0f), 0.999999f);
}

// ---------- kernel 4: heavy ragged sampling + per-block partial sums ----------
__global__ void sample_kernel(const float* __restrict__ edgeF,
                              const int*   __restrict__ begin,
                              const int*   __restrict__ totalp,
                              const float* __restrict__ rgb1,
                              const float* __restrict__ rgb2,
                              float* __restrict__ partials,
                              int N) {
    const int total = *totalp;
    float acc = 0.0f;
    for (int x = blockIdx.x * blockDim.x + threadIdx.x; x < total;
         x += gridDim.x * blockDim.x) {
        // binary search: largest e with begin[e] <= x
        int lo = 0, hi = N - 1;
        while (lo < hi) {
            int mid = (lo + hi + 1) >> 1;
            if (begin[mid] <= x) lo = mid; else hi = mid - 1;
        }
        const float* E = edgeF + (size_t)lo * EDGE_STRIDE;
        float cx = (float)(x - begin[lo]) * E[18];

        float u1x = E[0]*cx + E[6], u1y = E[1]*cx + E[7], u1z = E[2]*cx + E[8];
        float b1x = E[3], b1y = E[4], b1z = E[5];
        float u2x = E[9]*cx + E[15], u2y = E[10]*cx + E[16], u2z = E[11]*cx + E[17];
        float b2x = E[12], b2y = E[13], b2z = E[14];

#pragma unroll
        for (int v = 0; v < 10; ++v) {
            float cy = (float)v * (0.5f / 9.0f);
            float h1x = u1x + b1x*cy, h1y = u1y + b1y*cy, h1z = u1z + b1z*cy;
            float h2x = u2x + b2x*cy, h2y = u2y + b2y*cy, h2z = u2z + b2z*cy;
            float s1[3], s2[3];
            sample3(rgb1, clipdiv(h1x, h1z), clipdiv(h1y, h1z), s1);
            sample3(rgb2, clipdiv(h2x, h2z), clipdiv(h2y, h2z), s2);
            float d0 = s1[0]-s2[0], d1 = s1[1]-s2[1], d2 = s1[2]-s2[2];
            acc += d0*d0 + d1*d1 + d2*d2;
        }
    }
    __shared__ float red[SAMPLE_THREADS];
    red[threadIdx.x] = acc;
    __syncthreads();
    for (int s = SAMPLE_THREADS >> 1; s > 0; s >>= 1) {
        if (threadIdx.x < s) red[threadIdx.x] += red[threadIdx.x + s];
        __syncthreads();
    }
    if (threadIdx.x == 0) partials[blockIdx.x] = red[0];
}

// ---------- kernel 5: one-wave WMMA reduction ----------
// B = ones(4x16)  =>  D[m][n] = C[m][n] + sum_k A[m][k]; each WMMA folds 64 floats.
// Main loop is unconditional (8B-aligned pair loads -> global_load_b64, no EXEC
// juggling); one masked tail iteration handles n % 64.
__device__ float wmma_sum(const float* __restrict__ p, int n) {
    const int lane = (int)threadIdx.x;           // 0..31, one wave32
    const int m  = lane & 15;
    const int kb = (lane >> 4) << 1;             // 0 or 2
    const int li = m * 4 + kb;                   // lane's element offset in chunk
    v8f c = {0.f,0.f,0.f,0.f,0.f,0.f,0.f,0.f};
    v2f b; b.x = 1.0f; b.y = 1.0f;               // ones: layout-independent
    const int nfull = n & ~63;
    for (int base = 0; base < nfull; base += 64) {
        v2f a;
        a.x = p[base + li];
        a.y = p[base + li + 1];
        c = __builtin_amdgcn_wmma_f32_16x16x4_f32(
                false, a, false, b, (short)0, c, false, false);
    }
    if (nfull < n) {
        int i0 = nfull + li;
        v2f a;
        a.x = (i0     < n) ? p[i0]     : 0.0f;
        a.y = (i0 + 1 < n) ? p[i0 + 1] : 0.0f;
        c = __builtin_amdgcn_wmma_f32_16x16x4_f32(
                false, a, false, b, (short)0, c, false, false);
    }
    // lanes 0..15 hold row-sums m=0..7, lanes 16..31 hold m=8..15 (all n equal)
    float s = c[0]+c[1]+c[2]+c[3]+c[4]+c[5]+c[6]+c[7];
    s += __shfl_xor(s, 16, 32);
    return s;
}

__global__ void reduce_kernel(const float* __restrict__ partials, int nb,
                              const float* __restrict__ nl,
                              const float* __restrict__ sn, int N,
                              const int* __restrict__ totalp,
                              float* __restrict__ out) {
    float simSum = wmma_sum(partials, nb);
    float nlSum  = wmma_sum(nl, N);
    float snSum  = wmma_sum(sn, N);
    if (threadIdx.x == 0) {
        float total = (float)(*totalp);
        out[0] = simSum / (total * 10.0f * 3.0f);  // mean over (total_x*NUM_V, 3)
        out[1] = nlSum / (2.0f * (float)N);
        out[2] = snSum / (float)N;
    }
}

// ---------- host-side launch ----------
extern "C" void kernel_launch(void* const* d_in, const int* in_sizes, int n_in,
                              void* d_out, int out_size, void* d_ws, size_t ws_size,
                              hipStream_t stream) {
    const float* ep  = (const float*)d_in[0];
    const float* K1  = (const float*)d_in[1];
    const float* K2  = (const float*)d_in[2];
    const float* E1  = (const float*)d_in[3];
    const float* E2  = (const float*)d_in[4];
    const float* rgb1 = (const float*)d_in[5];
    const float* rgb2 = (const float*)d_in[6];
    const int N = in_sizes[0] / 12;

    float* ws       = (float*)d_ws;
    float* Tm       = ws;                         // 16
    float* edgeF    = Tm + 16;                    // EDGE_STRIDE * N
    float* nl       = edgeF + (size_t)EDGE_STRIDE * N;  // N
    float* sn       = nl + N;                     // N
    float* partials = sn + N;                     // SAMPLE_BLOCKS
    int*   numh     = (int*)(partials + SAMPLE_BLOCKS); // N
    int*   begin    = numh + N;                   // N
    int*   total    = begin + N;                  // 1

    precompute_T<<<1, 1, 0, stream>>>(K2, E1, E2, Tm);
    edge_kernel<<<(N + 255) / 256, 256, 0, stream>>>(ep, K1, Tm, edgeF, numh, nl, sn, N);
    scan_kernel<<<1, 1024, 0, stream>>>(numh, begin, total, N);
    sample_kernel<<<SAMPLE_BLOCKS, SAMPLE_THREADS, 0, stream>>>(
        edgeF, begin, total, rgb1, rgb2, partials, N);
    reduce_kernel<<<1, 32, 0, stream>>>(partials, SAMPLE_BLOCKS, nl, sn, N, total,
                                        (float*)d_out);
}